// Multi_Head_Attention_84129819394504
// MI455X (gfx1250) — compile-verified
//
#include <hip/hip_runtime.h>
#include <math.h>

// Problem constants (from reference): B=64, N=1024, D=128, M=8, DK=16
#define B_  64
#define N_  1024
#define D_  128
#define DK_ 16

typedef float v2f __attribute__((ext_vector_type(2)));
typedef float v8f __attribute__((ext_vector_type(8)));

// D = A(16x4 f32) * B(4x16 f32) + C(16x16 f32), wave32 WMMA
__device__ __forceinline__ v8f wmma4(v2f a, v2f b, v8f c) {
    return __builtin_amdgcn_wmma_f32_16x16x4_f32(
        /*neg_a=*/false, a, /*neg_b=*/false, b,
        /*c_mod=*/(short)0, c, /*reuse_a=*/false, /*reuse_b=*/false);
}

// ---------------------------------------------------------------------------
// Kernel 0: Wc[128,128] = Wv_flat[128,128] @ Wo_flat[128,128]
// 64 (16x16) tiles; 8 waves/block, 8 blocks; 32 k-chunks of 4.
// ---------------------------------------------------------------------------
__global__ void mha_wc_kernel(const float* __restrict__ Wv,
                              const float* __restrict__ Wo,
                              float* __restrict__ Wc) {
    const int wave = threadIdx.x >> 5;
    const int lane = threadIdx.x & 31;
    const int t     = blockIdx.x * 8 + wave;   // 0..63
    const int rbase = (t >> 3) * 16;
    const int cbase = (t & 7) * 16;
    const int n = lane & 15, h = lane >> 4;

    v8f acc = {};
#pragma unroll
    for (int cc = 0; cc < 32; ++cc) {
        const int k = 4 * cc + 2 * h;
        v2f a = *(const v2f*)(Wv + (rbase + n) * 128 + k);
        v2f b;
        b.x = Wo[(k    ) * 128 + cbase + n];
        b.y = Wo[(k + 1) * 128 + cbase + n];
        acc = wmma4(a, b, acc);
    }
#pragma unroll
    for (int g = 0; g < 8; ++g)
        Wc[(rbase + g + 8 * h) * 128 + cbase + n] = acc[g];
}

// ---------------------------------------------------------------------------
// Kernel 1: fused projections.  X:[B*N,128]
//   K = X@Wk (col tile 0), Q = X@Wq (tile 1), V' = X@Wc (tiles 2..9)
// One wave per 16-row tile; X A-chunks cached in registers across all 10
// column tiles (X read from HBM exactly once).
// ---------------------------------------------------------------------------
__global__ void mha_proj_kernel(const float* __restrict__ X,
                                const float* __restrict__ Wk,
                                const float* __restrict__ Wq,
                                const float* __restrict__ Wc,
                                float* __restrict__ Kp,
                                float* __restrict__ Qp,
                                float* __restrict__ Vp) {
    const int wave = threadIdx.x >> 5;
    const int lane = threadIdx.x & 31;
    const int rt    = blockIdx.x * 8 + wave;   // 0..4095 row tiles of B*N
    const int rbase = rt * 16;
    const int n = lane & 15, h = lane >> 4;

    // Cache 16x128 X tile in A-matrix layout: 32 chunks of float2
    v2f xa[32];
#pragma unroll
    for (int cc = 0; cc < 32; ++cc)
        xa[cc] = *(const v2f*)(X + (size_t)(rbase + n) * 128 + 4 * cc + 2 * h);

    for (int c = 0; c < 10; ++c) {
        const float* W; int ncols, colbase, dstride; float* dst;
        if (c == 0)      { W = Wk; ncols = 16;  colbase = 0;            dst = Kp; dstride = 16;  }
        else if (c == 1) { W = Wq; ncols = 16;  colbase = 0;            dst = Qp; dstride = 16;  }
        else             { W = Wc; ncols = 128; colbase = (c - 2) * 16; dst = Vp; dstride = 128; }

        v8f acc = {};
#pragma unroll
        for (int cc = 0; cc < 32; ++cc) {
            const int k = 4 * cc + 2 * h;
            v2f b;
            b.x = W[(k    ) * ncols + colbase + n];
            b.y = W[(k + 1) * ncols + colbase + n];
            acc = wmma4(xa[cc], b, acc);
        }
#pragma unroll
        for (int g = 0; g < 8; ++g)
            dst[(size_t)(rbase + g + 8 * h) * dstride + colbase + n] = acc[g];
    }
}

// ---------------------------------------------------------------------------
// Kernel 2: flash-style attention.
//   S[i,j] = K_i . Q_j * 0.25 * (1-mask)^(-1e10); P = softmax_j(S); O = P @ V'
// One wave per 16-row tile (4096 tiles); online softmax over 64 j-tiles.
// ---------------------------------------------------------------------------
__global__ void mha_attn_kernel(const float* __restrict__ Kp,
                                const float* __restrict__ Qp,
                                const float* __restrict__ Vp,
                                const float* __restrict__ mask,
                                float* __restrict__ out) {
    __shared__ float pbuf[8][16][17];          // padded: D-layout -> A-layout relay
    const int wave = threadIdx.x >> 5;
    const int lane = threadIdx.x & 31;
    const int tile  = blockIdx.x * 8 + wave;   // 0..4095
    const int b     = tile >> 6;
    const int ibase = (tile & 63) * 16;
    const int n = lane & 15, h = lane >> 4;

    const float* Kb = Kp + (size_t)b * N_ * DK_;
    const float* Qb = Qp + (size_t)b * N_ * DK_;
    const float* Vb = Vp + (size_t)b * N_ * D_;
    const float* Mb = mask + (size_t)b * N_ * N_;

    // K row-tile in A layout (loop invariant)
    v2f ka[4];
#pragma unroll
    for (int cc = 0; cc < 4; ++cc)
        ka[cc] = *(const v2f*)(Kb + (ibase + n) * DK_ + 4 * cc + 2 * h);

    float mrun[8], lrun[8];
    v8f o[8];
    const v8f vzero = {};
#pragma unroll
    for (int g = 0; g < 8; ++g) { mrun[g] = -3.0e38f; lrun[g] = 0.0f; }
#pragma unroll
    for (int dt = 0; dt < 8; ++dt) o[dt] = vzero;

    for (int jt = 0; jt < 64; ++jt) {
        const int jbase = jt * 16;

        // S tile = K_tile @ Q_tile^T  (inner dim 16 = 4 WMMA chunks)
        v8f s = {};
#pragma unroll
        for (int cc = 0; cc < 4; ++cc) {
            v2f qb = *(const v2f*)(Qb + (jbase + n) * DK_ + 4 * cc + 2 * h);
            s = wmma4(ka[cc], qb, s);
        }

        // scale + reference's pow-mask trick
#pragma unroll
        for (int g = 0; g < 8; ++g) {
            float mval = Mb[(size_t)(ibase + g + 8 * h) * N_ + jbase + n];
            float f = __powf(1.0f - mval, -1.0e10f);
            s[g] = s[g] * 0.25f * f;
        }

        // online softmax (row = g + 8h lives across the 16-lane half-group)
        float alpha[8];
#pragma unroll
        for (int g = 0; g < 8; ++g) {
            float tm = s[g];
            tm = fmaxf(tm, __shfl_xor(tm, 1, 32));
            tm = fmaxf(tm, __shfl_xor(tm, 2, 32));
            tm = fmaxf(tm, __shfl_xor(tm, 4, 32));
            tm = fmaxf(tm, __shfl_xor(tm, 8, 32));
            float mnew = fmaxf(mrun[g], tm);
            alpha[g] = __expf(mrun[g] - mnew);
            mrun[g] = mnew;
            float p = __expf(s[g] - mnew);
            s[g] = p;
            float ts = p;
            ts += __shfl_xor(ts, 1, 32);
            ts += __shfl_xor(ts, 2, 32);
            ts += __shfl_xor(ts, 4, 32);
            ts += __shfl_xor(ts, 8, 32);
            lrun[g] = lrun[g] * alpha[g] + ts;
        }

        // P: C/D layout -> LDS -> reload in A layout
#pragma unroll
        for (int g = 0; g < 8; ++g)
            pbuf[wave][g + 8 * h][n] = s[g];
        asm volatile("s_wait_dscnt 0" ::: "memory");
        v2f pa[4];
#pragma unroll
        for (int cc = 0; cc < 4; ++cc) {
            pa[cc].x = pbuf[wave][n][4 * cc + 2 * h];
            pa[cc].y = pbuf[wave][n][4 * cc + 2 * h + 1];
        }
        asm volatile("s_wait_dscnt 0" ::: "memory");

        // O = O*alpha + P @ V' (8 d-tiles x 4 chunks; V' hits L2)
#pragma unroll
        for (int dt = 0; dt < 8; ++dt) {
            v8f oo = o[dt];
#pragma unroll
            for (int g = 0; g < 8; ++g) oo[g] *= alpha[g];
#pragma unroll
            for (int cc = 0; cc < 4; ++cc) {
                const int j0 = jbase + 4 * cc + 2 * h;
                v2f vb;
                vb.x = Vb[(size_t)(j0    ) * D_ + dt * 16 + n];
                vb.y = Vb[(size_t)(j0 + 1) * D_ + dt * 16 + n];
                oo = wmma4(pa[cc], vb, oo);
            }
            o[dt] = oo;
        }
    }

    float inv[8];
#pragma unroll
    for (int g = 0; g < 8; ++g) inv[g] = 1.0f / lrun[g];
#pragma unroll
    for (int dt = 0; dt < 8; ++dt)
#pragma unroll
        for (int g = 0; g < 8; ++g)
            out[((size_t)b * N_ + ibase + g + 8 * h) * D_ + dt * 16 + n] =
                o[dt][g] * inv[g];
}

// ---------------------------------------------------------------------------
extern "C" void kernel_launch(void* const* d_in, const int* in_sizes, int n_in,
                              void* d_out, int out_size, void* d_ws, size_t ws_size,
                              hipStream_t stream) {
    const float* ref      = (const float*)d_in[0];  // [B,N,D]
    const float* mask     = (const float*)d_in[1];  // [B,N,N]
    const float* w_key    = (const float*)d_in[2];  // [D,DK]
    const float* w_query  = (const float*)d_in[3];  // [D,DK]
    const float* w_values = (const float*)d_in[4];  // [D,M,DK] == [128,128] flat
    const float* w_O      = (const float*)d_in[5];  // [M,DK,D] == [128,128] flat
    float* out = (float*)d_out;                     // [B,N,D] fp32

    // Workspace layout (floats): Wc[128*128] | K[B*N*16] | Q[B*N*16] | V'[B*N*128]
    float* ws = (float*)d_ws;
    float* Wc = ws;
    float* Kp = Wc + 128 * 128;
    float* Qp = Kp + (size_t)B_ * N_ * DK_;
    float* Vp = Qp + (size_t)B_ * N_ * DK_;

    // 0) Wc = Wv_flat @ Wo_flat  (fuses output projection into V)
    mha_wc_kernel<<<8, 256, 0, stream>>>(w_values, w_O, Wc);
    // 1) K/Q/V' projections (4096 row tiles, 8 waves/block)
    mha_proj_kernel<<<512, 256, 0, stream>>>(ref, w_key, w_query, Wc, Kp, Qp, Vp);
    // 2) attention (4096 row tiles, 8 waves/block)
    mha_attn_kernel<<<512, 256, 0, stream>>>(Kp, Qp, Vp, mask, out);
}